// MeanAggregator_89146341196166
// MI455X (gfx1250) — compile-verified
//
#include <hip/hip_runtime.h>

typedef float v2f __attribute__((ext_vector_type(2)));
typedef float v8f __attribute__((ext_vector_type(8)));

#define NN 1024
#define DD 64

// ---------------------------------------------------------------------------
// Stage 1: masked neighbor mean.  Bandwidth-bound: streams 512MB once.
// One block per (node, part).  16 m-groups x 16 float4 columns.
// ---------------------------------------------------------------------------
__global__ __launch_bounds__(256) void mean_agg_kernel(
    const float* __restrict__ adj,
    const float* __restrict__ neigh_real,
    const float* __restrict__ neigh_imag,
    float* __restrict__ mean_out /* [2][NN][DD] */)
{
    const int n    = blockIdx.x;
    const int part = blockIdx.y;
    const float* __restrict__ neigh = part ? neigh_imag : neigh_real;
    const float* __restrict__ base  = neigh + (size_t)n * NN * DD;
    const float* __restrict__ arow  = adj   + (size_t)n * NN;

    const int t  = threadIdx.x;
    const int d4 = t & 15;   // float4 column index (d = d4*4 .. d4*4+3)
    const int mg = t >> 4;   // m-group 0..15

    float4 acc = make_float4(0.f, 0.f, 0.f, 0.f);
    float  deg = 0.f;

#pragma unroll 4
    for (int m = mg; m < NN; m += 16) {
        const float  a = arow[m];
        const float4 v = *reinterpret_cast<const float4*>(base + (size_t)m * DD + d4 * 4);
        acc.x += a * v.x;
        acc.y += a * v.y;
        acc.z += a * v.z;
        acc.w += a * v.w;
        deg   += a;
    }

    __shared__ float4 lds[16][16];
    __shared__ float  ldsdeg[16];
    lds[mg][d4] = acc;
    if (d4 == 0) ldsdeg[mg] = deg;
    __syncthreads();

#pragma unroll
    for (int s = 8; s > 0; s >>= 1) {
        if (mg < s) {
            float4 o = lds[mg + s][d4];
            float4 c = lds[mg][d4];
            c.x += o.x; c.y += o.y; c.z += o.z; c.w += o.w;
            lds[mg][d4] = c;
            if (d4 == 0) ldsdeg[mg] += ldsdeg[mg + s];
        }
        __syncthreads();
    }

    if (mg == 0) {
        const float dsum = ldsdeg[0];
        float4 s = lds[0][d4];
        float4 r;
        r.x = s.x / dsum; r.y = s.y / dsum; r.z = s.z / dsum; r.w = s.w / dsum;
        if (!__builtin_isfinite(r.x)) r.x = 0.f;
        if (!__builtin_isfinite(r.y)) r.y = 0.f;
        if (!__builtin_isfinite(r.z)) r.z = 0.f;
        if (!__builtin_isfinite(r.w)) r.w = 0.f;
        *reinterpret_cast<float4*>(
            mean_out + ((size_t)part * NN + n) * DD + d4 * 4) = r;
    }
}

// ---------------------------------------------------------------------------
// Stage 2: out = relu(self + mean @ W) via V_WMMA_F32_16X16X4_F32.
// One wave32 per 16x16 output tile; K=64 swept in 16 K=4 WMMA steps.
// A(16x4):  lane L -> M = L&15, reg r -> K = (L>>4)*2 + r
// B(4x16):  lane L -> N = L&15, reg r -> K = (L>>4)*2 + r
// C/D:      lane L, reg r -> M = (L>>4)*8 + r, N = L&15
// ---------------------------------------------------------------------------
__global__ __launch_bounds__(128) void gemm_self_relu_kernel(
    const float* __restrict__ mean,       // [2][NN][DD]
    const float* __restrict__ W,          // [DD][DD]
    const float* __restrict__ self_real,  // [NN][DD]
    const float* __restrict__ self_imag,  // [NN][DD]
    float* __restrict__ out)              // [2][NN][DD]
{
    const int lane = threadIdx.x & 31;
    const int wave = threadIdx.x >> 5;
    const int w    = blockIdx.x * 4 + wave;     // 0..511 tiles
    const int part = w >> 8;                    // 256 tiles per part
    const int rem  = w & 255;
    const int n0   = (rem >> 2) * 16;           // 64 n-tiles
    const int d0   = (rem & 3)  * 16;           // 4 d-tiles

    const float* __restrict__ meanp = mean + (size_t)part * NN * DD;
    const float* __restrict__ selfp = part ? self_imag : self_real;

    const int row = lane & 15;            // M for A-frag, N for B/D-frags
    const int kb  = (lane >> 4) * 2;      // K sub-offset within the K=4 step

    v8f acc = {0.f, 0.f, 0.f, 0.f, 0.f, 0.f, 0.f, 0.f};

#pragma unroll
    for (int kk = 0; kk < DD; kk += 4) {
        v2f a, b;
        const float* ap = meanp + (size_t)(n0 + row) * DD + (kk + kb);
        a.x = ap[0];
        a.y = ap[1];
        b.x = W[(size_t)(kk + kb) * DD + d0 + row];
        b.y = W[(size_t)(kk + kb + 1) * DD + d0 + row];
        acc = __builtin_amdgcn_wmma_f32_16x16x4_f32(
            /*neg_a=*/false, a, /*neg_b=*/false, b,
            /*c_mod=*/(short)0, acc, /*reuse_a=*/false, /*reuse_b=*/false);
    }

#pragma unroll
    for (int r = 0; r < 8; ++r) {
        const int m  = (lane >> 4) * 8 + r;
        const int nn = n0 + m;
        const int dd = d0 + row;
        float v = acc[r] + selfp[(size_t)nn * DD + dd];
        v = v > 0.f ? v : 0.f;
        out[(size_t)part * NN * DD + (size_t)nn * DD + dd] = v;
    }
}

// ---------------------------------------------------------------------------
extern "C" void kernel_launch(void* const* d_in, const int* in_sizes, int n_in,
                              void* d_out, int out_size, void* d_ws, size_t ws_size,
                              hipStream_t stream) {
    (void)in_sizes; (void)n_in; (void)out_size; (void)ws_size;

    const float* adj = (const float*)d_in[0];
    const float* nr  = (const float*)d_in[1];
    const float* ni  = (const float*)d_in[2];
    const float* sr  = (const float*)d_in[3];
    const float* si  = (const float*)d_in[4];
    const float* W   = (const float*)d_in[5];

    float* mean = (float*)d_ws;     // 2*1024*64 floats = 512 KB scratch
    float* out  = (float*)d_out;    // out_r (1024x64) then out_i (1024x64)

    dim3 g1(NN, 2);
    mean_agg_kernel<<<g1, 256, 0, stream>>>(adj, nr, ni, mean);

    // 512 tiles, 4 waves (128 threads) per block -> 128 blocks
    gemm_self_relu_kernel<<<128, 128, 0, stream>>>(mean, W, sr, si, out);
}